// FuzzyLayer_10264971838088
// MI455X (gfx1250) — compile-verified
//
#include <hip/hip_runtime.h>
#include <cstdint>

// FuzzyLayer: out[b,s,d*256+i] = exp(-(x[b,s,i]-mu[d,i])^2 / sigma[d,i])
// B=16, S=2048, I=256, D=8.
// Traffic: read 33.6 MB (x) + write 268.4 MB (out) => ~13 us floor @ 23.3 TB/s.
// Pure streaming kernel: b128 NT loads/stores, params staged in LDS via the
// CDNA5 async global->LDS path, one v_exp_f32 per output element.

#define BATCH 16
#define SEQ   2048
#define ISZ   256
#define DEG   8
#define ROWS  (BATCH * SEQ)              // 32768
#define I4    (ISZ / 4)                  // 64 float4 per row
#define ROWS_PER_GROUP 4
#define NGROUPS (ROWS / ROWS_PER_GROUP)  // 8192 row-groups
#define THREADS 256
#define GRID    2048                     // persistent blocks; 4 groups each
#define LOG2E 1.4426950408889634f

typedef float f32x4 __attribute__((ext_vector_type(4)));

__global__ __launch_bounds__(THREADS) void fuzzy_kernel(
    const float* __restrict__ x,
    const float* __restrict__ params,   // (DEG*ISZ, 2): [2r]=mu, [2r+1]=sigma
    float* __restrict__ out)
{
    // LDS layout (floats): [0,2048) mu, [2048,4096) c = -log2e/sigma,
    //                      [4096,8192) raw staged params (16 KB).
    __shared__ __align__(16) float smem[8192];

    const int tid = threadIdx.x;
    const int sub = tid >> 6;   // row within group (0..3)
    const int li  = tid & 63;   // float4 index within row (0..63)

    // ---- CDNA5 async global->LDS staging of the 16 KB param table ----
    // 256 lanes x 16 B x 4 rounds = 16 KB. LDS byte address = low 32 bits of
    // the flat shared pointer (LDS aperture maps addr[31:0] -> LDS offset).
    {
        uint64_t g = (uint64_t)(uintptr_t)params + (uint32_t)tid * 16u;
        uint32_t l = (uint32_t)(uintptr_t)(&smem[4096]) + (uint32_t)tid * 16u;
#pragma unroll
        for (int r = 0; r < 4; ++r) {
            asm volatile("global_load_async_to_lds_b128 %0, %1, off"
                         :: "v"(l), "v"(g) : "memory");
            g += 4096u;
            l += 4096u;
        }
        asm volatile("s_wait_asynccnt 0" ::: "memory");
    }
    __syncthreads();

    // Transform (once per block): mu -> smem[0..2047],
    // c = -log2(e) * rcp(sigma) -> smem[2048..4095].  v_rcp_f32 (~1 ulp) is
    // plenty for a Gaussian membership function and avoids the IEEE div chain.
#pragma unroll
    for (int k = tid; k < DEG * ISZ; k += THREADS) {
        float mu = smem[4096 + 2 * k];
        float sg = smem[4096 + 2 * k + 1];
        smem[k]        = mu;
        smem[2048 + k] = -LOG2E * __builtin_amdgcn_rcpf(sg);
    }
    __syncthreads();

    const f32x4* __restrict__ xf4  = (const f32x4*)x;
    const f32x4* __restrict__ muf4 = (const f32x4*)&smem[0];
    const f32x4* __restrict__ cf4  = (const f32x4*)&smem[2048];
    f32x4* __restrict__ of4 = (f32x4*)out;

    // ---- Persistent grid-stride over row-groups ----
    for (int grp = blockIdx.x; grp < NGROUPS; grp += GRID) {
        const int row = grp * ROWS_PER_GROUP + sub;

        // Prefetch next iteration's x float4 (gfx1250 global_prefetch_b8,
        // no counter, hides the read behind this iteration's 8 stores).
        const int gnext = grp + GRID;
        if (gnext < NGROUPS) {
            __builtin_prefetch(
                &xf4[(gnext * ROWS_PER_GROUP + sub) * I4 + li], 0, 3);
        }

        f32x4 xv = __builtin_nontemporal_load(&xf4[row * I4 + li]);
        const int obase = row * (DEG * I4);   // row * 512 float4s

#pragma unroll
        for (int d = 0; d < DEG; ++d) {
            f32x4 m = muf4[d * I4 + li];
            f32x4 c = cf4[d * I4 + li];
            f32x4 o;
            float t;
            t = xv.x - m.x; o.x = __builtin_amdgcn_exp2f(t * t * c.x);
            t = xv.y - m.y; o.y = __builtin_amdgcn_exp2f(t * t * c.y);
            t = xv.z - m.z; o.z = __builtin_amdgcn_exp2f(t * t * c.z);
            t = xv.w - m.w; o.w = __builtin_amdgcn_exp2f(t * t * c.w);
            // Streaming store: written once, never re-read; bypass L2 dirty.
            __builtin_nontemporal_store(o, &of4[obase + d * I4 + li]);
        }
    }
}

extern "C" void kernel_launch(void* const* d_in, const int* in_sizes, int n_in,
                              void* d_out, int out_size, void* d_ws, size_t ws_size,
                              hipStream_t stream) {
    const float* x  = (const float*)d_in[0];
    const float* fp = (const float*)d_in[1];
    float* out = (float*)d_out;
    (void)in_sizes; (void)n_in; (void)out_size; (void)d_ws; (void)ws_size;

    hipLaunchKernelGGL(fuzzy_kernel, dim3(GRID), dim3(THREADS), 0, stream,
                       x, fp, out);
}